// GATLayer_84593675862505
// MI455X (gfx1250) — compile-verified
//
#include <hip/hip_runtime.h>
#include <hip/hip_bf16.h>
#include <math.h>

#define ALPHA 0.2f

typedef float v2f __attribute__((ext_vector_type(2)));
typedef float v8f __attribute__((ext_vector_type(8)));

// ---------------------------------------------------------------------------
// Kernel 1: h = X @ W via v_wmma_f32_16x16x4_f32.
// Each wave computes a 16-row x 64-col slab of h (4 accumulators of 16x16),
// looping K = 64 in steps of 4. W (64x64, 16KB) staged in LDS per block.
// ---------------------------------------------------------------------------
__global__ __launch_bounds__(256) void gat_gemm_wmma(
    const float* __restrict__ X, const float* __restrict__ W,
    float* __restrict__ H, int N)
{
    __shared__ float Wlds[64 * 64];

    // cooperative load of W into LDS: 4096 floats / 256 threads = 16 each
    for (int i = threadIdx.x; i < 64 * 64; i += 256) {
        Wlds[i] = W[i];
    }
    __syncthreads();

    const int wid  = threadIdx.x >> 5;          // wave in block (0..7)
    const int lane = threadIdx.x & 31;
    const int half = lane >> 4;                 // 0: lanes 0-15, 1: lanes 16-31
    const int l    = lane & 15;

    const int tile = blockIdx.x * 8 + wid;      // 16-row tile index
    const int nTiles = (N + 15) >> 4;
    if (tile >= nTiles) return;                 // wave-uniform: EXEC stays all-1s

    const int row0 = tile * 16;

    v8f c0 = {}, c1 = {}, c2 = {}, c3 = {};

    const float* xrow = X + (size_t)(row0 + l) * 64;

    #pragma unroll
    for (int k = 0; k < 64; k += 4) {
        // A fragment (16x4 f32): lane half selects K pair per ISA layout
        v2f a;
        a.x = xrow[k + 2 * half + 0];
        a.y = xrow[k + 2 * half + 1];

        // B fragments (4x16 f32) for the four 16-col output tiles
        #pragma unroll
        for (int t = 0; t < 4; ++t) {
            const int n = t * 16;
            v2f b;
            b.x = Wlds[(k + 2 * half + 0) * 64 + n + l];
            b.y = Wlds[(k + 2 * half + 1) * 64 + n + l];
            if (t == 0) c0 = __builtin_amdgcn_wmma_f32_16x16x4_f32(false, a, false, b, (short)0, c0, false, false);
            if (t == 1) c1 = __builtin_amdgcn_wmma_f32_16x16x4_f32(false, a, false, b, (short)0, c1, false, false);
            if (t == 2) c2 = __builtin_amdgcn_wmma_f32_16x16x4_f32(false, a, false, b, (short)0, c2, false, false);
            if (t == 3) c3 = __builtin_amdgcn_wmma_f32_16x16x4_f32(false, a, false, b, (short)0, c3, false, false);
        }
    }

    // Store: C/D layout -> lane gives column, VGPR r gives row (+8 for hi half)
    #pragma unroll
    for (int r = 0; r < 8; ++r) {
        const int row = row0 + half * 8 + r;
        float* hrow = H + (size_t)row * 64;
        hrow[ 0 + l] = c0[r];
        hrow[16 + l] = c1[r];
        hrow[32 + l] = c2[r];
        hrow[48 + l] = c3[r];
    }
}

// monotone float<->uint encoding for atomic max over signed floats
__device__ __forceinline__ unsigned int enc_f32(float f) {
    unsigned int u = __float_as_uint(f);
    return (u & 0x80000000u) ? ~u : (u | 0x80000000u);
}
__device__ __forceinline__ float dec_f32(unsigned int u) {
    u = (u & 0x80000000u) ? (u & 0x7FFFFFFFu) : ~u;
    return __uint_as_float(u);
}

// ---------------------------------------------------------------------------
// Kernel 2: per-node scores s_src = h.a[:64], s_tgt = h.a[64:], wave32
// reduction; also init m_enc = enc(-inf), denom = 0, and zero d_out.
// One wave per node, each lane handles 2 of 64 columns.
// ---------------------------------------------------------------------------
__global__ __launch_bounds__(256) void gat_scores_init(
    const float* __restrict__ H, const float* __restrict__ a,
    float* __restrict__ s_src, float* __restrict__ s_tgt,
    unsigned int* __restrict__ m_enc, float* __restrict__ denom,
    float* __restrict__ out, int N)
{
    const int node = blockIdx.x * 8 + (threadIdx.x >> 5);
    const int lane = threadIdx.x & 31;
    if (node >= N) return;

    const float h0 = H[(size_t)node * 64 + lane];
    const float h1 = H[(size_t)node * 64 + 32 + lane];

    float ss = h0 * a[lane]      + h1 * a[32 + lane];
    float st = h0 * a[64 + lane] + h1 * a[96 + lane];

    #pragma unroll
    for (int o = 16; o > 0; o >>= 1) {
        ss += __shfl_xor(ss, o, 32);
        st += __shfl_xor(st, o, 32);
    }

    out[(size_t)node * 64 + lane]      = 0.0f;
    out[(size_t)node * 64 + 32 + lane] = 0.0f;

    if (lane == 0) {
        s_src[node] = ss;
        s_tgt[node] = st;
        m_enc[node] = enc_f32(-INFINITY);
        denom[node] = 0.0f;
    }
}

// ---------------------------------------------------------------------------
// Kernel 3: edge scores e = leaky_relu(s_src[src]+s_tgt[tgt]); segment max
// via atomicMax on monotone-encoded u32.
// ---------------------------------------------------------------------------
__global__ __launch_bounds__(256) void gat_edge_max(
    const int* __restrict__ ei, const float* __restrict__ s_src,
    const float* __restrict__ s_tgt, float* __restrict__ e_ws,
    unsigned int* __restrict__ m_enc, int E)
{
    const int i = blockIdx.x * 256 + threadIdx.x;
    if (i >= E) return;
    const int s = ei[i];
    const int t = ei[E + i];
    float e = s_src[s] + s_tgt[t];
    e = (e > 0.0f) ? e : ALPHA * e;
    e_ws[i] = e;
    atomicMax(&m_enc[t], enc_f32(e));
}

// ---------------------------------------------------------------------------
// Kernel 4: p = exp(e - m[tgt]); segment sum via atomicAdd; store p in place.
// ---------------------------------------------------------------------------
__global__ __launch_bounds__(256) void gat_edge_expsum(
    const int* __restrict__ ei, float* __restrict__ e_ws,
    const unsigned int* __restrict__ m_enc, float* __restrict__ denom, int E)
{
    const int i = blockIdx.x * 256 + threadIdx.x;
    if (i >= E) return;
    const int t = ei[E + i];
    const float p = expf(e_ws[i] - dec_f32(m_enc[t]));
    e_ws[i] = p;
    atomicAdd(&denom[t], p);
}

// ---------------------------------------------------------------------------
// Kernel 5: weighted scatter-add. One wave per edge, 2 columns per lane:
// out[tgt][c] += h[src][c] * (p / denom[tgt])
// ---------------------------------------------------------------------------
__global__ __launch_bounds__(256) void gat_edge_scatter(
    const int* __restrict__ ei, const float* __restrict__ e_ws,
    const float* __restrict__ denom, const float* __restrict__ H,
    float* __restrict__ out, int E)
{
    const int edge = blockIdx.x * 8 + (threadIdx.x >> 5);
    const int lane = threadIdx.x & 31;
    if (edge >= E) return;

    const int s = ei[edge];
    const int t = ei[E + edge];
    const float att = e_ws[edge] / denom[t];

    const float v0 = H[(size_t)s * 64 + lane]      * att;
    const float v1 = H[(size_t)s * 64 + 32 + lane] * att;
    atomicAdd(&out[(size_t)t * 64 + lane],      v0);
    atomicAdd(&out[(size_t)t * 64 + 32 + lane], v1);
}

// ---------------------------------------------------------------------------
// Kernel 6: in-place ELU on d_out.
// ---------------------------------------------------------------------------
__global__ __launch_bounds__(256) void gat_elu(float* __restrict__ out, int n)
{
    const int i = blockIdx.x * 256 + threadIdx.x;
    if (i >= n) return;
    const float v = out[i];
    out[i] = (v > 0.0f) ? v : (expf(v) - 1.0f);
}

// ---------------------------------------------------------------------------
extern "C" void kernel_launch(void* const* d_in, const int* in_sizes, int n_in,
                              void* d_out, int out_size, void* d_ws, size_t ws_size,
                              hipStream_t stream)
{
    const float* X  = (const float*)d_in[0];   // [N,64]
    const int*   EI = (const int*)d_in[1];     // [2,E]
    const float* W  = (const float*)d_in[2];   // [64,64]
    const float* a  = (const float*)d_in[3];   // [128]
    float* out      = (float*)d_out;           // [N,64]

    const int N = in_sizes[0] / 64;
    const int E = in_sizes[1] / 2;

    // workspace carve-up (256B aligned)
    auto align256 = [](size_t x) { return (x + 255) & ~(size_t)255; };
    char* base = (char*)d_ws;
    size_t off = 0;
    float* H            = (float*)(base + off); off = align256(off + (size_t)N * 64 * sizeof(float));
    float* s_src        = (float*)(base + off); off = align256(off + (size_t)N * sizeof(float));
    float* s_tgt        = (float*)(base + off); off = align256(off + (size_t)N * sizeof(float));
    unsigned int* m_enc = (unsigned int*)(base + off); off = align256(off + (size_t)N * sizeof(unsigned int));
    float* denom        = (float*)(base + off); off = align256(off + (size_t)N * sizeof(float));
    float* e_ws         = (float*)(base + off); off = align256(off + (size_t)E * sizeof(float));

    // 1) WMMA GEMM: 8 waves/block, 16 rows per wave -> 128 rows per block
    {
        const int nTiles = (N + 15) / 16;
        const int blocks = (nTiles + 7) / 8;
        gat_gemm_wmma<<<blocks, 256, 0, stream>>>(X, W, H, N);
    }
    // 2) node scores + init
    {
        const int blocks = (N + 7) / 8;
        gat_scores_init<<<blocks, 256, 0, stream>>>(H, a, s_src, s_tgt, m_enc, denom, out, N);
    }
    // 3) edge leaky-relu + segment max
    {
        const int blocks = (E + 255) / 256;
        gat_edge_max<<<blocks, 256, 0, stream>>>(EI, s_src, s_tgt, e_ws, m_enc, E);
    }
    // 4) exp + segment sum
    {
        const int blocks = (E + 255) / 256;
        gat_edge_expsum<<<blocks, 256, 0, stream>>>(EI, e_ws, m_enc, denom, E);
    }
    // 5) weighted scatter-add (one wave per edge)
    {
        const int blocks = (E + 7) / 8;
        gat_edge_scatter<<<blocks, 256, 0, stream>>>(EI, e_ws, denom, H, out, E);
    }
    // 6) in-place ELU
    {
        const int n = N * 64;
        const int blocks = (n + 255) / 256;
        gat_elu<<<blocks, 256, 0, stream>>>(out, n);
    }
}